// HistogramLoss_70549132804802
// MI455X (gfx1250) — compile-verified
//
#include <hip/hip_runtime.h>
#include <hip/hip_bf16.h>
#include <math.h>

#ifndef __has_builtin
#define __has_builtin(x) 0
#endif

// gfx1250 async global->LDS copy path (ASYNCcnt-tracked). Guarded so the file
// compiles on toolchains that lack the builtins (falls back to sync copy).
#if __has_builtin(__builtin_amdgcn_global_load_async_to_lds_b128)
#define ASYNC_LDS 1
#else
#define ASYNC_LDS 0
#endif

#define NB 64
#define BLK 256
#define NWAVE (BLK / 32)
#define NSUB 4                 // sub-histogram copies per wave (lane & 3)
#define CSTRIDE 132            // 2*64 bins + 4 pad dwords -> bank-skewed copies
#define NCOPY (NWAVE * NSUB)   // 32 copies per block
#define GRID_BLOCKS 2048

typedef int v4i __attribute__((ext_vector_type(4)));
typedef __attribute__((address_space(1))) v4i* gas_v4i;
typedef __attribute__((address_space(3))) v4i* las_v4i;

// ---------- ordered-uint encoding of f32 (monotonic, for integer atomics) ----------
__device__ __forceinline__ unsigned f2ord(float f) {
  unsigned u = __float_as_uint(f);
  return (u & 0x80000000u) ? ~u : (u | 0x80000000u);
}
__device__ __forceinline__ float ord2f(unsigned u) {
  return __uint_as_float((u & 0x80000000u) ? (u & 0x7FFFFFFFu) : ~u);
}

// ---------- async copy helpers ----------
__device__ __forceinline__ void async_cp16(float4* lds_dst, const float4* gsrc) {
#if ASYNC_LDS
  __builtin_amdgcn_global_load_async_to_lds_b128(
      (gas_v4i)(v4i*)(void*)gsrc,
      (las_v4i)(v4i*)(void*)lds_dst,
      /*imm offset*/ 0, /*cpol: RT, keep resident in L2 for 2nd pass*/ 0);
#else
  *lds_dst = *gsrc;
#endif
}

template <int N>
__device__ __forceinline__ void wait_async() {
#if ASYNC_LDS
#if __has_builtin(__builtin_amdgcn_s_wait_asynccnt)
  __builtin_amdgcn_s_wait_asynccnt(N);
#else
  asm volatile("s_wait_asynccnt %0" ::"n"(N) : "memory");
#endif
  asm volatile("" ::: "memory");  // compiler barrier: LDS reads stay below the wait
#endif
}

// ---------- workspace layout (in d_ws, re-initialized every launch) ----------
// ws[0] = ordered-uint min slot, ws[1] = ordered-uint max slot, ws[2..129] = 2x64 bins
__global__ void k_init(unsigned* __restrict__ ws) {
  int t = threadIdx.x;
  if (t == 0) ws[0] = 0xFFFFFFFFu;       // +inf in ordered encoding (min identity)
  else if (t < 130) ws[t] = 0u;          // ws[1]=0 => -inf ordered (max identity); bins=0
}

// ---------- pass 1: global min/max (HBM-stream bound: ~128MB @ 23.3TB/s) ----------
__global__ void k_minmax(const float* __restrict__ a, const float* __restrict__ b,
                         int n, unsigned* __restrict__ mm) {
  __shared__ float smn[NWAVE], smx[NWAVE];
  const int tid = threadIdx.x;
  const int gid = blockIdx.x * BLK + tid;
  const int gstride = gridDim.x * BLK;
  const int n4 = n >> 2;

  float mn = INFINITY, mx = -INFINITY;
  const float4* a4 = (const float4*)a;
  const float4* b4 = (const float4*)b;
  for (int i = gid; i < n4; i += gstride) {
    float4 v = a4[i];
    mn = fminf(mn, fminf(fminf(v.x, v.y), fminf(v.z, v.w)));
    mx = fmaxf(mx, fmaxf(fmaxf(v.x, v.y), fmaxf(v.z, v.w)));
  }
  for (int i = gid; i < n4; i += gstride) {
    float4 v = b4[i];
    mn = fminf(mn, fminf(fminf(v.x, v.y), fminf(v.z, v.w)));
    mx = fmaxf(mx, fmaxf(fmaxf(v.x, v.y), fmaxf(v.z, v.w)));
  }
  // scalar tail (n not multiple of 4)
  int tail0 = n4 << 2;
  if (gid < n - tail0) {
    float x = a[tail0 + gid], y = b[tail0 + gid];
    mn = fminf(mn, fminf(x, y));
    mx = fmaxf(mx, fmaxf(x, y));
  }

  // wave32 reduction
  for (int off = 16; off; off >>= 1) {
    mn = fminf(mn, __shfl_xor(mn, off, 32));
    mx = fmaxf(mx, __shfl_xor(mx, off, 32));
  }
  if ((tid & 31) == 0) { smn[tid >> 5] = mn; smx[tid >> 5] = mx; }
  __syncthreads();
  if (tid == 0) {
    for (int w = 1; w < NWAVE; ++w) { mn = fminf(mn, smn[w]); mx = fmaxf(mx, smx[w]); }
    atomicMin(&mm[0], f2ord(mn));
    atomicMax(&mm[1], f2ord(mx));
  }
}

// ---------- binning helpers ----------
__device__ __forceinline__ void bin_add(unsigned* hist, float x, float minv, float rng) {
  // exact reference order: floor(((x - min) / rng) * 64), clamp to [0,63]
  float t = (x - minv) / rng * 64.0f;
  int bi = (int)floorf(t);
  bi = bi < 0 ? 0 : (bi > (NB - 1) ? (NB - 1) : bi);
  atomicAdd(&hist[bi], 1u);  // ds_add_u32 (no return)
}

__device__ __forceinline__ void hist_stream(const float4* __restrict__ src, int n4,
                                            int gbase, int gstride, float minv, float rng,
                                            unsigned* hist, float4* s0, float4* s1) {
  int i = gbase;
  if (i < n4) async_cp16(s0, &src[i]);  // prologue
  int buf = 0;
  while (i < n4) {
    int nx = i + gstride;
    if (nx < n4) {
      async_cp16(buf ? s0 : s1, &src[nx]);  // prefetch next tile
      wait_async<1>();                      // current tile resident
    } else {
      wait_async<0>();                      // drain
    }
    float4 v = buf ? *s1 : *s0;             // ds_load_b128, own slot -> no barrier
    bin_add(hist, v.x, minv, rng);
    bin_add(hist, v.y, minv, rng);
    bin_add(hist, v.z, minv, rng);
    bin_add(hist, v.w, minv, rng);
    i = nx;
    buf ^= 1;
  }
}

// ---------- pass 2: histograms (lane-replicated LDS hists, async-staged loads) ----------
__global__ void k_hist(const float* __restrict__ a, const float* __restrict__ b,
                       int n, const unsigned* __restrict__ mm, unsigned* __restrict__ gbins) {
  __shared__ float4 stage[2][BLK];             // 8 KB double-buffer staging
  __shared__ unsigned whist[NCOPY * CSTRIDE];  // ~17 KB: 32 bank-skewed sub-histograms

  const int tid = threadIdx.x;
  const int wave = tid >> 5;
  const int copy = wave * NSUB + (tid & (NSUB - 1));  // 8 lanes share a copy
  unsigned* myhist = &whist[copy * CSTRIDE];          // [0..63]=tensor a, [64..127]=tensor b

  for (int i = tid; i < NCOPY * CSTRIDE; i += BLK) whist[i] = 0u;
  __syncthreads();

  const float minv = ord2f(mm[0]);
  const float maxv = ord2f(mm[1]);
  float rng = maxv - minv;
  if (rng == 0.0f) rng = 1.0f;

  const int n4 = n >> 2;
  const int gstride = gridDim.x * BLK;
  const int gbase = blockIdx.x * BLK + tid;

  hist_stream((const float4*)a, n4, gbase, gstride, minv, rng,
              myhist, &stage[0][tid], &stage[1][tid]);
  hist_stream((const float4*)b, n4, gbase, gstride, minv, rng,
              myhist + NB, &stage[0][tid], &stage[1][tid]);

  // scalar tail
  int tail0 = n4 << 2;
  if (gbase < n - tail0) {
    bin_add(myhist, a[tail0 + gbase], minv, rng);
    bin_add(myhist + NB, b[tail0 + gbase], minv, rng);
  }

  __syncthreads();
  if (tid < 2 * NB) {
    unsigned s = 0;
    for (int c = 0; c < NCOPY; ++c) s += whist[c * CSTRIDE + tid];
    if (s) atomicAdd(&gbins[tid], s);
  }
}

// ---------- final: mean(|h_a - h_b|) over 64 bins ----------
__global__ void k_final(const unsigned* __restrict__ gbins, float* __restrict__ out) {
  __shared__ float s[NB];
  const int t = threadIdx.x;  // 64 threads
  s[t] = fabsf((float)gbins[t] - (float)gbins[NB + t]);
  __syncthreads();
  for (int off = NB / 2; off; off >>= 1) {
    if (t < off) s[t] += s[t + off];
    __syncthreads();
  }
  if (t == 0) out[0] = s[0] * (1.0f / 64.0f);
}

extern "C" void kernel_launch(void* const* d_in, const int* in_sizes, int n_in,
                              void* d_out, int out_size, void* d_ws, size_t ws_size,
                              hipStream_t stream) {
  const float* a = (const float*)d_in[0];
  const float* b = (const float*)d_in[1];
  const int n = in_sizes[0];  // both tensors same shape
  unsigned* ws = (unsigned*)d_ws;
  unsigned* mm = ws;          // [0]=min(ord), [1]=max(ord)
  unsigned* gbins = ws + 2;   // 2 x 64 bins
  float* out = (float*)d_out;

  int blocks = GRID_BLOCKS;
  int n4 = n >> 2;
  if (n4 > 0 && n4 < blocks * BLK) blocks = (n4 + BLK - 1) / BLK;
  if (blocks < 1) blocks = 1;

  k_init<<<1, 160, 0, stream>>>(ws);
  k_minmax<<<blocks, BLK, 0, stream>>>(a, b, n, mm);
  k_hist<<<blocks, BLK, 0, stream>>>(a, b, n, mm, gbins);
  k_final<<<1, NB, 0, stream>>>(gbins, out);
}